// M_Attention_10866267259042
// MI455X (gfx1250) — compile-verified
//
#include <hip/hip_runtime.h>
#include <hip/hip_bf16.h>
#include <math.h>

// ---------------------------------------------------------------------------
// Additive attention (Bahdanau-style) for B=64, T=512, E2=1024, DEC=1024.
//
// Memory-bound: en_output = 134 MB f32, streamed twice (score + context)
// => ~12 us floor at 23.3 TB/s. All GEMMs run on V_WMMA_F32_16X16X4_F32
// (native f32 WMMA, no precision change needed since traffic dominates).
// ---------------------------------------------------------------------------

typedef __attribute__((ext_vector_type(2))) float v2f;
typedef __attribute__((ext_vector_type(8))) float v8f;

#define BATCH 64
#define TLEN  512
#define E2DIM 1024
#define DECD  1024

// ---------------------------------------------------------------------------
// Kernel 1: s0 = tanh(s @ W0 + b0)   [64,1024]
// One wave (32 threads) per 16x16 output tile: 4 row-tiles x 64 col-tiles.
// K = 1024 in steps of 4 with v_wmma_f32_16x16x4_f32.
// ---------------------------------------------------------------------------
__global__ __launch_bounds__(32) void
attn_s0_kernel(const float* __restrict__ s, const float* __restrict__ W0,
               const float* __restrict__ b0, float* __restrict__ s0out) {
    const int tile   = blockIdx.x;         // 0..255
    const int rt     = tile >> 6;          // 0..3   (16-row tile of B)
    const int ct     = tile & 63;          // 0..63  (16-col tile of E2)
    const int L      = threadIdx.x;        // 0..31
    const int half   = L >> 4;             // 0: K=k0,k0+1   1: K=k0+2,k0+3
    const int lane16 = L & 15;
    const int row    = rt * 16 + lane16;   // A-matrix M index (ISA 7.12.2)
    const int n      = ct * 16 + lane16;   // B/C/D N index

    v8f c = {};
    for (int k0 = 0; k0 < DECD; k0 += 4) {
        const int k = k0 + half * 2;
        v2f a = *(const v2f*)(s + (size_t)row * DECD + k);      // A 16x4 frag
        v2f b;                                                   // B 4x16 frag
        b.x = W0[(size_t)k * E2DIM + n];
        b.y = W0[(size_t)(k + 1) * E2DIM + n];
        c = __builtin_amdgcn_wmma_f32_16x16x4_f32(false, a, false, b,
                                                  (short)0, c, false, false);
    }
    const float bias  = b0[n];
    const int   mbase = rt * 16 + half * 8;        // C/D layout: VGPR v -> M=v(+8)
    for (int v = 0; v < 8; ++v)
        s0out[(size_t)(mbase + v) * E2DIM + n] = tanhf(c[v] + bias);
}

// ---------------------------------------------------------------------------
// Kernel 2: s1 = s0 @ W1b + b1       [64,16]  (W1b = W1 rows 1024..2047,
// zero-padded from N=10 to N=16 in LDS). One workgroup, 4 waves.
// ---------------------------------------------------------------------------
__global__ __launch_bounds__(128) void
attn_s1_kernel(const float* __restrict__ s0, const float* __restrict__ W1,
               const float* __restrict__ b1, float* __restrict__ s1out) {
    __shared__ float ldsB[DECD * 16];              // 64 KB (WGP LDS is 320 KB)
    const int tid = threadIdx.x;                   // 0..127
    for (int i = tid; i < DECD * 16; i += 128) {
        const int k = i >> 4, n = i & 15;
        ldsB[i] = (n < 10) ? W1[(size_t)(DECD + k) * 10 + n] : 0.f;
    }
    __syncthreads();

    const int wave = tid >> 5;                     // row tile 0..3
    const int L = tid & 31, half = L >> 4, lane16 = L & 15;
    const int row = wave * 16 + lane16;

    v8f c = {};
    for (int k0 = 0; k0 < DECD; k0 += 4) {
        const int k = k0 + half * 2;
        v2f a = *(const v2f*)(s0 + (size_t)row * E2DIM + k);
        v2f b;
        b.x = ldsB[k * 16 + lane16];
        b.y = ldsB[(k + 1) * 16 + lane16];
        c = __builtin_amdgcn_wmma_f32_16x16x4_f32(false, a, false, b,
                                                  (short)0, c, false, false);
    }
    const float bias  = (lane16 < 10) ? b1[lane16] : 0.f;
    const int   mbase = wave * 16 + half * 8;
    for (int v = 0; v < 8; ++v)
        s1out[(mbase + v) * 16 + lane16] = c[v] + bias;
}

// ---------------------------------------------------------------------------
// Kernel 3 (dominant, streams en_output once):
//   logit[b,t] = tanh(en[b,t,:] @ W1a + s1[b,:]) . W2 + b2
// Grid: 64 batches x 8 tile-groups; 4 waves/WG, 16 T-rows per wave.
// W1a zero-padded [1024][16] cached in LDS per workgroup.
// ---------------------------------------------------------------------------
__global__ __launch_bounds__(128) void
attn_score_kernel(const float* __restrict__ en, const float* __restrict__ W1,
                  const float* __restrict__ s1, const float* __restrict__ W2,
                  const float* __restrict__ b2, float* __restrict__ logits) {
    __shared__ float ldsB[E2DIM * 16];             // 64 KB
    const int b   = blockIdx.x >> 3;               // batch
    const int tg  = blockIdx.x & 7;                // tile group (4 tiles each)
    const int tid = threadIdx.x;                   // 0..127

    for (int i = tid; i < E2DIM * 16; i += 128) {
        const int k = i >> 4, n = i & 15;
        ldsB[i] = (n < 10) ? W1[(size_t)k * 10 + n] : 0.f;
    }
    __syncthreads();

    const int wave = tid >> 5, L = tid & 31, half = L >> 4, lane16 = L & 15;
    const int t0  = (tg * 4 + wave) * 16;
    const int row = t0 + lane16;
    const float* arow = en + ((size_t)b * TLEN + row) * E2DIM;

    v8f c = {};
    for (int k0 = 0; k0 < E2DIM; k0 += 4) {
        const int k = k0 + half * 2;
        if ((k0 & 31) == 0)                          // gfx1250 global_prefetch_b8
            __builtin_prefetch(arow + k0 + 128, 0, 0);
        v2f a = *(const v2f*)(arow + k);             // 8B load, sequential per row
        v2f bf;
        bf.x = ldsB[k * 16 + lane16];
        bf.y = ldsB[(k + 1) * 16 + lane16];
        c = __builtin_amdgcn_wmma_f32_16x16x4_f32(false, a, false, bf,
                                                  (short)0, c, false, false);
    }

    // Epilogue: + s1, tanh, dot with W2 (N<10), reduce over N within each
    // 16-lane half (C layout: lanes 0-15 -> M=v, lanes 16-31 -> M=v+8).
    const float s1v   = s1[b * 16 + lane16];         // cols 10..15 are exact 0
    const float w2v   = (lane16 < 10) ? W2[lane16] : 0.f;
    const float bias2 = b2[0];
    for (int v = 0; v < 8; ++v) {
        float t = tanhf(c[v] + s1v) * w2v;
        for (int m = 8; m >= 1; m >>= 1)
            t += __shfl_xor(t, m, 16);
        if (lane16 == 0)
            logits[(size_t)b * TLEN + t0 + half * 8 + v] = t + bias2;
    }
}

// ---------------------------------------------------------------------------
// Kernel 4: softmax over T + context[b,e] = sum_t alpha[t] * en[b,t,e].
// One workgroup per batch, 256 threads, float4-coalesced streaming of en.
// ---------------------------------------------------------------------------
__global__ __launch_bounds__(256) void
attn_context_kernel(const float* __restrict__ en, const float* __restrict__ logits,
                    float* __restrict__ out) {
    __shared__ float alpha[TLEN];
    __shared__ float red[256];
    const int b = blockIdx.x, tid = threadIdx.x;

    const float l0 = logits[(size_t)b * TLEN + tid];
    const float l1 = logits[(size_t)b * TLEN + 256 + tid];

    red[tid] = fmaxf(l0, l1);
    __syncthreads();
    for (int s = 128; s > 0; s >>= 1) {
        if (tid < s) red[tid] = fmaxf(red[tid], red[tid + s]);
        __syncthreads();
    }
    const float mx = red[0];
    __syncthreads();

    const float e0 = __expf(l0 - mx), e1 = __expf(l1 - mx);
    alpha[tid] = e0; alpha[tid + 256] = e1;
    red[tid] = e0 + e1;
    __syncthreads();
    for (int s = 128; s > 0; s >>= 1) {
        if (tid < s) red[tid] += red[tid + s];
        __syncthreads();
    }
    const float inv = 1.f / red[0];
    __syncthreads();
    alpha[tid] *= inv; alpha[tid + 256] *= inv;
    __syncthreads();

    // Each thread owns 4 consecutive e-columns -> 4 KB fully-coalesced row loads.
    float4 acc = {0.f, 0.f, 0.f, 0.f};
    const float4* base = (const float4*)(en + (size_t)b * TLEN * E2DIM) + tid;
    for (int t = 0; t < TLEN; ++t) {
        const float a = alpha[t];
        const float4 x = base[(size_t)t * (E2DIM / 4)];
        acc.x += a * x.x; acc.y += a * x.y; acc.z += a * x.z; acc.w += a * x.w;
    }
    ((float4*)(out + (size_t)b * E2DIM))[tid] = acc;
}

// ---------------------------------------------------------------------------
extern "C" void kernel_launch(void* const* d_in, const int* in_sizes, int n_in,
                              void* d_out, int out_size, void* d_ws, size_t ws_size,
                              hipStream_t stream) {
    const float* en = (const float*)d_in[0];   // [64,512,1024]
    const float* s  = (const float*)d_in[1];   // [64,1024]
    const float* W0 = (const float*)d_in[2];   // [1024,1024]
    const float* b0 = (const float*)d_in[3];   // [1024]
    const float* W1 = (const float*)d_in[4];   // [2048,10]
    const float* b1 = (const float*)d_in[5];   // [10]
    const float* W2 = (const float*)d_in[6];   // [10,1]
    const float* b2 = (const float*)d_in[7];   // [1]
    float* out = (float*)d_out;                // [64,1,1024]

    // Workspace layout (f32): s0 [64*1024] | s1 [64*16] | logits [64*512]
    float* s0w     = (float*)d_ws;
    float* s1w     = s0w + BATCH * E2DIM;
    float* logitsw = s1w + BATCH * 16;

    attn_s0_kernel     <<<256, 32,  0, stream>>>(s, W0, b0, s0w);
    attn_s1_kernel     <<<1,   128, 0, stream>>>(s0w, W1, b1, s1w);
    attn_score_kernel  <<<512, 128, 0, stream>>>(en, W1, s1w, W2, b2, logitsw);
    attn_context_kernel<<<64,  256, 0, stream>>>(en, logitsw, out);
}